// IntermediateWeigher_19164144074936
// MI455X (gfx1250) — compile-verified
//
#include <hip/hip_runtime.h>

#define NUM_LAYERS 12
#define EMBED_DIM  768
#define N_OUT      4
#define BT_TOTAL   (4 * 2048)   // B*T = 8192 tokens
#define EPS        1e-5f

typedef __attribute__((ext_vector_type(2))) float v2f;
typedef __attribute__((ext_vector_type(8))) float v8f;

// One block per token (b,t). 256 threads = 8 waves (wave32).
// Wave owns 96 columns = 6 groups of 16. Lane mapping chosen to match the
// V_WMMA_F32_16X16X4_F32 B-matrix layout: lane half h, VGPR j <-> K = 2h+j.
__launch_bounds__(256)
__global__ void ln_layermix_wmma_kernel(const float* __restrict__ x,
                                        const float* __restrict__ gamma,
                                        const float* __restrict__ beta,
                                        const float* __restrict__ w,
                                        const float* __restrict__ bias,
                                        float* __restrict__ out)
{
    __shared__ float s_sum[NUM_LAYERS];
    __shared__ float s_sq[NUM_LAYERS];
    __shared__ float s_mean[NUM_LAYERS];
    __shared__ float s_rstd[NUM_LAYERS];

    const int tid  = threadIdx.x;
    const int bt   = blockIdx.x;          // token index (b*2048 + t)
    const int lane = tid & 31;
    const int wv   = tid >> 5;            // wave id 0..7
    const int hv   = lane >> 4;           // half of wave (0/1) -> K pairing
    const int li   = lane & 15;           // column-in-group / M row

    if (tid < NUM_LAYERS) { s_sum[tid] = 0.0f; s_sq[tid] = 0.0f; }
    __syncthreads();

    // ---- single-pass load: 36 fp32 per lane, kept in registers ----
    // vals[k][p][g] = x[layer = 4k + 2*hv + p][bt][c0 + 16*g]
    const int c0 = wv * 96 + li;
    float vals[3][2][6];
    #pragma unroll
    for (int k = 0; k < 3; ++k) {
        #pragma unroll
        for (int p = 0; p < 2; ++p) {
            const int l = 4 * k + 2 * hv + p;
            const float* xp = x + ((size_t)l * BT_TOTAL + bt) * EMBED_DIM + c0;
            #pragma unroll
            for (int g = 0; g < 6; ++g)
                vals[k][p][g] = xp[g * 16];
        }
    }

    // ---- per-layer mean / var: lane sums -> 16-lane butterfly -> LDS atomics ----
    #pragma unroll
    for (int k = 0; k < 3; ++k) {
        #pragma unroll
        for (int p = 0; p < 2; ++p) {
            float s = 0.0f, q = 0.0f;
            #pragma unroll
            for (int g = 0; g < 6; ++g) {
                const float v = vals[k][p][g];
                s += v;
                q += v * v;
            }
            // xor masks 1..8 stay inside the 16-lane half (bit4 untouched)
            #pragma unroll
            for (int m = 1; m <= 8; m <<= 1) {
                s += __shfl_xor(s, m, 32);
                q += __shfl_xor(q, m, 32);
            }
            if (li == 0) {
                const int l = 4 * k + 2 * hv + p;
                atomicAdd(&s_sum[l], s);   // ds_add_f32
                atomicAdd(&s_sq[l],  q);
            }
        }
    }
    __syncthreads();

    if (tid < NUM_LAYERS) {
        const float mean = s_sum[tid] * (1.0f / EMBED_DIM);
        const float var  = s_sq[tid] * (1.0f / EMBED_DIM) - mean * mean;
        s_mean[tid] = mean;
        s_rstd[tid] = rsqrtf(var + EPS);
    }
    __syncthreads();

    // ---- normalize in registers with folded affine ----
    #pragma unroll
    for (int k = 0; k < 3; ++k) {
        #pragma unroll
        for (int p = 0; p < 2; ++p) {
            const int l = 4 * k + 2 * hv + p;
            const float mean = s_mean[l];
            const float rstd = s_rstd[l];
            const float* gp = gamma + l * EMBED_DIM + c0;
            const float* bp = beta  + l * EMBED_DIM + c0;
            #pragma unroll
            for (int g = 0; g < 6; ++g) {
                const float ga = gp[g * 16] * rstd;            // gamma * rstd
                const float bb = fmaf(-mean, ga, bp[g * 16]);  // beta - mean*ga
                vals[k][p][g] = fmaf(vals[k][p][g], ga, bb);
            }
        }
    }

    // ---- A matrices (16x4 per K-chunk): row m = li, K = 4k + 2*hv + j ----
    v2f a[3];
    #pragma unroll
    for (int k = 0; k < 3; ++k) {
        float a0 = 0.0f, a1 = 0.0f;
        if (li < N_OUT) {
            a0 = w[li * NUM_LAYERS + 4 * k + 2 * hv + 0];
            a1 = w[li * NUM_LAYERS + 4 * k + 2 * hv + 1];
        }
        a[k][0] = a0;
        a[k][1] = a1;
    }
    const float bv0 = bias[0], bv1 = bias[1], bv2 = bias[2], bv3 = bias[3];

    // ---- per 16-column group: 3 chained V_WMMA_F32_16X16X4_F32 (K=12) ----
    #pragma unroll
    for (int g = 0; g < 6; ++g) {
        v2f b0, b1, b2;
        b0[0] = vals[0][0][g]; b0[1] = vals[0][1][g];
        b1[0] = vals[1][0][g]; b1[1] = vals[1][1][g];
        b2[0] = vals[2][0][g]; b2[1] = vals[2][1][g];

        v8f acc = { bv0, bv1, bv2, bv3, 0.0f, 0.0f, 0.0f, 0.0f };
        acc = __builtin_amdgcn_wmma_f32_16x16x4_f32(false, a[0], false, b0,
                                                    (short)0, acc, false, false);
        acc = __builtin_amdgcn_wmma_f32_16x16x4_f32(false, a[1], false, b1,
                                                    (short)0, acc, false, false);
        acc = __builtin_amdgcn_wmma_f32_16x16x4_f32(false, a[2], false, b2,
                                                    (short)0, acc, false, false);

        // D rows 0..3 live in acc[0..3] on lanes 0..15 (N = lane)
        if (lane < 16) {
            const int c = wv * 96 + g * 16 + li;
            #pragma unroll
            for (int n = 0; n < N_OUT; ++n)
                out[((size_t)n * BT_TOTAL + bt) * EMBED_DIM + c] = acc[n];
        }
    }
}

extern "C" void kernel_launch(void* const* d_in, const int* in_sizes, int n_in,
                              void* d_out, int out_size, void* d_ws, size_t ws_size,
                              hipStream_t stream) {
    const float* x     = (const float*)d_in[0];
    const float* gamma = (const float*)d_in[1];
    const float* beta  = (const float*)d_in[2];
    const float* w     = (const float*)d_in[3];
    const float* b     = (const float*)d_in[4];
    float* out = (float*)d_out;

    dim3 grid(BT_TOTAL);   // 8192 tokens
    dim3 block(256);       // 8 wave32 waves
    hipLaunchKernelGGL(ln_layermix_wmma_kernel, grid, block, 0, stream,
                       x, gamma, beta, w, b, out);
}